// VariableGroupNorm_24867860644004
// MI455X (gfx1250) — compile-verified
//
#include <hip/hip_runtime.h>
#include <hip/hip_bf16.h>

#define N_  32
#define C_  256
#define HW_ 3136
#define G_  16

typedef float v2f __attribute__((ext_vector_type(2)));
typedef float v8f __attribute__((ext_vector_type(8)));

// ---------------------------------------------------------------------------
// Kernel 0: build seg[p] = group index of permuted channel position p.
// ---------------------------------------------------------------------------
__global__ void seg_kernel(const int* __restrict__ gsizes, int* __restrict__ seg) {
    if (threadIdx.x == 0) {
        int c = 0;
        for (int g = 0; g < G_; ++g) {
            int s = gsizes[g];
            for (int i = 0; i < s; ++i) seg[c++] = g;
        }
    }
}

// ---------------------------------------------------------------------------
// Kernel 1: per-row (n,c) sum and sum-of-squares over HW = 3136 floats.
// One wave per row; streams data through LDS with CDNA5 async loads
// (global_load_async_to_lds_b128) using a 2-deep double buffer per wave.
// 3136 = 24 * 128 + 64 : 24 b128 iterations + a 2-float scalar tail per lane.
// ---------------------------------------------------------------------------
__global__ __launch_bounds__(256) void rowstats_kernel(const float* __restrict__ x,
                                                       float* __restrict__ rowsum,
                                                       float* __restrict__ rowsq) {
    __shared__ float lbuf[8][2][128];   // 8 waves x double buffer x 128 floats = 8 KB
    const int lane = threadIdx.x & 31;
    const int wave = threadIdx.x >> 5;
    const int row  = blockIdx.x * 8 + wave;
    const float* rp = x + (size_t)row * HW_;

    unsigned lds0 = (unsigned)(uintptr_t)&lbuf[wave][0][lane * 4];
    unsigned lds1 = (unsigned)(uintptr_t)&lbuf[wave][1][lane * 4];

    float s = 0.0f, q = 0.0f;

    // prologue: issue iteration 0 into buffer 0
    {
        unsigned long long ga = (unsigned long long)(uintptr_t)(rp + lane * 4);
        asm volatile("global_load_async_to_lds_b128 %0, %1, off"
                     :: "v"(lds0), "v"(ga) : "memory");
    }
    for (int it = 0; it < 24; ++it) {
        if (it + 1 < 24) {
            unsigned nlds = ((it + 1) & 1) ? lds1 : lds0;
            unsigned long long ga =
                (unsigned long long)(uintptr_t)(rp + (it + 1) * 128 + lane * 4);
            asm volatile("global_load_async_to_lds_b128 %0, %1, off"
                         :: "v"(nlds), "v"(ga) : "memory");
            asm volatile("s_wait_asynccnt 0x1" ::: "memory");
        } else {
            asm volatile("s_wait_asynccnt 0x0" ::: "memory");
        }
        const float4 v = *(const float4*)&lbuf[wave][it & 1][lane * 4];
        s += (v.x + v.y) + (v.z + v.w);
        q += (v.x * v.x + v.y * v.y) + (v.z * v.z + v.w * v.w);
    }
    // tail: 64 floats, 2 per lane, direct global loads
    {
        float a0 = rp[3072 + lane * 2];
        float a1 = rp[3072 + lane * 2 + 1];
        s += a0 + a1;
        q += a0 * a0 + a1 * a1;
    }
    // wave32 butterfly reduction
    for (int off = 16; off > 0; off >>= 1) {
        s += __shfl_xor(s, off, 32);
        q += __shfl_xor(q, off, 32);
    }
    if (lane == 0) {
        rowsum[row] = s;
        rowsq[row]  = q;
    }
}

// ---------------------------------------------------------------------------
// Kernel 2: segment-sum of row stats into groups via v_wmma_f32_16x16x4_f32.
// One wave per sample n. D = sum_k A(16x4) * B(4x16) accumulated over 64
// K-chunks of 4 permuted channel positions:
//   A[g][k] = (seg[chunk*4+k] == g) ? 1 : 0        (group indicator)
//   B[k][0] = rowsum[indexes[n*C + chunk*4+k]]
//   B[k][1] = rowsq [indexes[n*C + chunk*4+k]]     (other columns 0)
// => D[g][0] = group sum, D[g][1] = group sumsq, all in f32.
// A layout (16x4 f32): lane L holds row M=L%16; VGPR0 = K (L<16 ? 0 : 2),
// VGPR1 = K (L<16 ? 1 : 3).  B/D rows striped across lanes within a VGPR.
// All gathers are staged into LDS up-front so the WMMA loop is straight-line
// (ds_load + cndmask/fma + wmma), no predicated global loads.
// ---------------------------------------------------------------------------
__global__ __launch_bounds__(32) void groupstats_kernel(const float* __restrict__ rowsum,
                                                        const float* __restrict__ rowsq,
                                                        const int*  __restrict__ indexes,
                                                        const int*  __restrict__ seg,
                                                        const int*  __restrict__ gsizes,
                                                        float* __restrict__ mu,
                                                        float* __restrict__ ivar) {
    const int n    = blockIdx.x;
    const int lane = threadIdx.x;
    const int m    = lane & 15;     // row (A) / column (B,D) index
    const int hi   = lane >> 4;     // second K half
    const int base = n * C_;

    // Stage permuted per-channel stats + segment map into LDS.
    __shared__ float sbuf[C_];
    __shared__ float qbuf[C_];
    __shared__ int   segl[C_];
    #pragma unroll
    for (int p = lane; p < C_; p += 32) {
        const int idx = indexes[base + p];
        sbuf[p] = rowsum[idx];
        qbuf[p] = rowsq[idx];
        segl[p] = seg[p];
    }
    __syncthreads();

    // Loop-invariant arithmetic column selectors (avoid branchy selects).
    const float wsel_s = (m == 0) ? 1.0f : 0.0f;
    const float wsel_q = (m == 1) ? 1.0f : 0.0f;
    const int   k0off  = hi ? 2 : 0;
    const int   k1off  = hi ? 3 : 1;

    v8f c = {0.f, 0.f, 0.f, 0.f, 0.f, 0.f, 0.f, 0.f};

    for (int chunk = 0; chunk < 64; ++chunk) {
        const int p0 = chunk * 4 + k0off;
        const int p1 = chunk * 4 + k1off;

        v2f a, b;
        a.x = (segl[p0] == m) ? 1.0f : 0.0f;
        a.y = (segl[p1] == m) ? 1.0f : 0.0f;
        b.x = fmaf(sbuf[p0], wsel_s, qbuf[p0] * wsel_q);
        b.y = fmaf(sbuf[p1], wsel_s, qbuf[p1] * wsel_q);

        c = __builtin_amdgcn_wmma_f32_16x16x4_f32(
                /*neg_a=*/false, a, /*neg_b=*/false, b,
                /*c_mod=*/(short)0, c, /*reuse_a=*/false, /*reuse_b=*/false);
    }

    // D layout: VGPR j holds M=j (lanes 0-15, N=lane) and M=j+8 (lanes 16-31).
    // Column N=0 is group sum, N=1 is group sumsq.
    __shared__ float gsum[G_], gsq[G_];
    #pragma unroll
    for (int j = 0; j < 8; ++j) {
        const int g = hi ? (j + 8) : j;
        if (m == 0) gsum[g] = c[j];
        if (m == 1) gsq[g]  = c[j];
    }
    __syncthreads();

    if (lane < G_) {
        const int g = lane;
        const float ne   = (float)gsizes[g] * (float)HW_;
        const float mean = gsum[g] / ne;
        const float var  = (gsq[g] - ne * mean * mean) / (ne - 1.0f);
        mu[n * G_ + g]   = mean;
        ivar[n * G_ + g] = rsqrtf(var + 1e-12f);
    }
}

// ---------------------------------------------------------------------------
// Kernel 3: normalize. out[j] = x[j]*scale + shift with
//   p = reverse_indexes[j] - n*C, g = seg[p],
//   scale = ivar[n,g]*weight[p], shift = bias[p] - mu[n,g]*scale.
// One wave per row; float4 streaming (x read should mostly hit 192MB L2).
// ---------------------------------------------------------------------------
__global__ __launch_bounds__(256) void normalize_kernel(const float* __restrict__ x,
                                                        const float* __restrict__ weight,
                                                        const float* __restrict__ bias,
                                                        const int*  __restrict__ rev,
                                                        const int*  __restrict__ seg,
                                                        const float* __restrict__ mu,
                                                        const float* __restrict__ ivar,
                                                        float* __restrict__ out) {
    const int lane = threadIdx.x & 31;
    const int wave = threadIdx.x >> 5;
    const int row  = blockIdx.x * 8 + wave;
    const int n    = row >> 8;                 // row / C_
    const int p    = rev[row] - n * C_;        // permuted position of this channel
    const int g    = seg[p];
    const float mn    = mu[n * G_ + g];
    const float iv    = ivar[n * G_ + g];
    const float scale = iv * weight[p];
    const float shift = bias[p] - mn * scale;

    const float4* rp = (const float4*)(x   + (size_t)row * HW_);
    float4*       op = (float4*)      (out + (size_t)row * HW_);

    #pragma unroll 4
    for (int it = 0; it < 24; ++it) {
        float4 v = rp[it * 32 + lane];
        v.x = fmaf(v.x, scale, shift);
        v.y = fmaf(v.y, scale, shift);
        v.z = fmaf(v.z, scale, shift);
        v.w = fmaf(v.w, scale, shift);
        op[it * 32 + lane] = v;
    }
    // tail: 64 floats
    {
        const size_t t = (size_t)row * HW_ + 3072 + lane * 2;
        out[t]     = fmaf(x[t],     scale, shift);
        out[t + 1] = fmaf(x[t + 1], scale, shift);
    }
}

// ---------------------------------------------------------------------------
extern "C" void kernel_launch(void* const* d_in, const int* in_sizes, int n_in,
                              void* d_out, int out_size, void* d_ws, size_t ws_size,
                              hipStream_t stream) {
    const float* x       = (const float*)d_in[0];
    const float* weight  = (const float*)d_in[1];
    const float* bias    = (const float*)d_in[2];
    const int*   gsizes  = (const int*)d_in[3];
    const int*   indexes = (const int*)d_in[4];
    const int*   rev     = (const int*)d_in[5];
    float*       out     = (float*)d_out;

    // workspace layout (floats): rowsum[8192] rowsq[8192] mu[512] ivar[512] seg[256]
    float* ws     = (float*)d_ws;
    float* rowsum = ws;
    float* rowsq  = ws + (N_ * C_);
    float* mu     = ws + 2 * (N_ * C_);
    float* ivar   = mu + (N_ * G_);
    int*   seg    = (int*)(ivar + (N_ * G_));

    seg_kernel<<<1, 32, 0, stream>>>(gsizes, seg);
    rowstats_kernel<<<(N_ * C_) / 8, 256, 0, stream>>>(x, rowsum, rowsq);
    groupstats_kernel<<<N_, 32, 0, stream>>>(rowsum, rowsq, indexes, seg, gsizes, mu, ivar);
    normalize_kernel<<<(N_ * C_) / 8, 256, 0, stream>>>(x, weight, bias, rev, seg, mu, ivar, out);
}